// Network_56556129354611
// MI455X (gfx1250) — compile-verified
//
#include <hip/hip_runtime.h>

// ---------------------------------------------------------------------------
// Spiking net (exc layer only; inh layer is dead code — its state is never
// returned by the reference). Weights for each 16-row group live in VGPRs for
// all 128 timesteps. Per step, one fused pass per wave:
//   w = clip(w + 1e-3*z[row]*t_pre[col] - 1e-3*t_post[row]*x[t,col], 0, 1)
//   C += WMMA_F32_16x16x4( A = w-pair (already in A layout), B = x[t+1] bcast )
// giving next step's 16 row-currents replicated in the accumulator (no
// cross-lane reduce). t_pre depends only on the input stream -> precomputed
// for all T by a small scan kernel into d_ws.
// ---------------------------------------------------------------------------

typedef __attribute__((ext_vector_type(2))) float v2f;
typedef __attribute__((ext_vector_type(8))) float v8f;

#define T_STEPS 128
#define N_IN    2048
#define N_EXC   2048
#define WAVES   16                 // waves per block
#define TILES   32                 // 4-column wmma tiles per wave (16*32*4 == 2048)
#define BLOCK   (WAVES * 32)

// LIF / STDP constants (DT=1e-3 folded in)
#define K_MEM   0.1f               // DT*TAU_MEM_INV
#define K_SYN   0.8f               // 1 - DT*TAU_SYN_INV
#define K_TR    0.05f              // DT*TAU_PRE_INV == DT*TAU_POST_INV
#define ETA     1e-3f
#define RHO_RST 5.0f

// --- kernel A: t_pre trace scan over T, scaled by ETA_PLUS ------------------
__global__ void tpre_scan_kernel(const float* __restrict__ x,
                                 float* __restrict__ tpre) {
    int k = blockIdx.x * blockDim.x + threadIdx.x;
    if (k >= N_IN) return;
    float tp = 0.0f;
    for (int t = 0; t < T_STEPS; ++t) {
        tp = (1.0f - K_TR) * tp + K_TR * x[t * N_IN + k];
        tpre[t * N_IN + k] = ETA * tp;             // store eta_plus * t_pre(t)
    }
}

// --- kernel B: persistent per-row-group SNN ---------------------------------
__launch_bounds__(BLOCK)
__global__ void snn_exc_kernel(const float* __restrict__ x,      // [T, N_IN]
                               const float* __restrict__ w_exc,  // [N_EXC, N_IN] (read-only)
                               const float* __restrict__ tpre,   // [T, N_IN], pre-scaled
                               float* __restrict__ spikes) {     // [T, N_EXC]
    __shared__ float part[WAVES][16];                  // per-wave partial dots
    __shared__ float vS[16], iS[16], rhoS[16], tpS[16];
    __shared__ float zcS[16], tcS[16];                 // broadcast update coefs

    const int tid    = threadIdx.x;
    const int lane   = tid & 31;
    const int wid    = tid >> 5;
    const int rowgrp = blockIdx.x;                     // 16 rows per block
    const int row    = (rowgrp << 4) + (lane & 15);    // my row (WMMA A: M = lane&15)
    const int colOff = wid * (TILES * 4) + ((lane >> 4) << 1); // A layout: hi half-wave K=2,3

    if (tid < 16) { vS[tid] = 0.f; iS[tid] = 0.f; rhoS[tid] = 0.f; tpS[tid] = 0.f; }

    // Load my weight slab into registers (never written back — not an output).
    v2f w[TILES];
    const float* wrow = w_exc + (size_t)row * N_IN + colOff;
#pragma unroll
    for (int it = 0; it < TILES; ++it)
        w[it] = *(const v2f*)(wrow + it * 4);

    // Priming pass: cur(0) = dot(x[0], we_initial), replicated in C.
    v8f c;
#pragma unroll
    for (int r = 0; r < 8; ++r) c[r] = 0.0f;
    {
        const float* xp = x + colOff;
#pragma unroll
        for (int it = 0; it < TILES; ++it) {
            v2f b = *(const v2f*)(xp + it * 4);    // B[k,n] = x[base+k] for all n
            c = __builtin_amdgcn_wmma_f32_16x16x4_f32(
                    false, w[it], false, b, (short)0, c, false, false);
        }
    }
    __syncthreads();                                   // also covers state init

    for (int t = 0; t < T_STEPS; ++t) {
        // Publish this wave's 16 partial row-dots (lane0: rows 0-7, lane16: rows 8-15).
        if (lane == 0) {
#pragma unroll
            for (int r = 0; r < 8; ++r) part[wid][r] = c[r];
        } else if (lane == 16) {
#pragma unroll
            for (int r = 0; r < 8; ++r) part[wid][8 + r] = c[r];
        }
        __syncthreads();

        if (tid < 16) {
            float cur = 0.0f;
#pragma unroll
            for (int ww = 0; ww < WAVES; ++ww) cur += part[ww][tid];

            // LIF with refractory state (matches reference ordering exactly).
            float v = vS[tid], ii = iS[tid], rho = rhoS[tid], tpost = tpS[tid];
            float v_dec  = v + K_MEM * (ii - v);
            float i_new  = K_SYN * ii + cur;
            float z      = (v_dec > 1.0f) ? 1.0f : 0.0f;
            float refrac = (rho > 0.0f) ? 1.0f : 0.0f;
            float v_new  = ((z != 0.0f) || (refrac != 0.0f)) ? 0.0f : v_dec;
            z            = (refrac != 0.0f) ? 0.0f : z;
            float rho_n  = (z != 0.0f) ? RHO_RST : fmaxf(rho - refrac, 0.0f);
            tpost        = (1.0f - K_TR) * tpost + K_TR * z;   // trace update first

            vS[tid] = v_new; iS[tid] = i_new; rhoS[tid] = rho_n; tpS[tid] = tpost;
            zcS[tid] = z;                  // dw += z * (eta*t_pre[k])  (tpre pre-scaled)
            tcS[tid] = -ETA * tpost;       // dw += (-eta*t_post) * x[t,k]
            spikes[t * N_EXC + (rowgrp << 4) + tid] = z;
        }
        __syncthreads();
        if (t == T_STEPS - 1) break;       // final weights are never observed

        // Fused pass: STDP-update my weights with step-t terms, and accumulate
        // next step's dot against x[t+1] via WMMA on the freshly updated pair.
        const float zc = zcS[lane & 15];
        const float tc = tcS[lane & 15];
        const float* xt = x    + (size_t)t * N_IN + colOff;
        const float* xn = xt + N_IN;
        const float* tp = tpre + (size_t)t * N_IN + colOff;
#pragma unroll
        for (int r = 0; r < 8; ++r) c[r] = 0.0f;
#pragma unroll
        for (int it = 0; it < TILES; ++it) {
            v2f xt2 = *(const v2f*)(xt + it * 4);
            v2f xn2 = *(const v2f*)(xn + it * 4);
            v2f tp2 = *(const v2f*)(tp + it * 4);
            v2f wn;
            wn.x = fminf(fmaxf(fmaf(zc, tp2.x, fmaf(tc, xt2.x, w[it].x)), 0.0f), 1.0f);
            wn.y = fminf(fmaxf(fmaf(zc, tp2.y, fmaf(tc, xt2.y, w[it].y)), 0.0f), 1.0f);
            w[it] = wn;
            c = __builtin_amdgcn_wmma_f32_16x16x4_f32(
                    false, wn, false, xn2, (short)0, c, false, false);
        }
    }
}

extern "C" void kernel_launch(void* const* d_in, const int* in_sizes, int n_in,
                              void* d_out, int out_size, void* d_ws, size_t ws_size,
                              hipStream_t stream) {
    (void)in_sizes; (void)n_in; (void)out_size; (void)ws_size;
    const float* x     = (const float*)d_in[0];   // exc_currents [T, N_IN]
    const float* w_exc = (const float*)d_in[1];   // [N_EXC, N_IN]
    // d_in[2] (w_inh) intentionally unused: inhibitory layer does not affect output.
    float* spikes = (float*)d_out;                // [T, N_EXC]
    float* tpre   = (float*)d_ws;                 // T*N_IN floats (1 MB)

    tpre_scan_kernel<<<(N_IN + 255) / 256, 256, 0, stream>>>(x, tpre);
    snn_exc_kernel<<<N_EXC / 16, BLOCK, 0, stream>>>(x, w_exc, tpre, spikes);
}